// SpERT_50268297232850
// MI455X (gfx1250) — compile-verified
//
#include <hip/hip_runtime.h>
#include <hip/hip_bf16.h>

typedef __attribute__((ext_vector_type(2))) float v2f;
typedef __attribute__((ext_vector_type(8))) float v8f;

#define B_   4
#define S_   256
#define H_   768
#define E_   200
#define R_   200
#define D_   25
#define NET  9
#define NRT  5
#define KSPAN 1561           // D + 2H (reference)
#define KREL  2354           // 2D + 3H (reference)
#define DP_   32             // padded size-emb stride (cols 25..31 = 0)
#define KSP   (2*H_ + DP_)   // 1568: padded span K (multiple of 4)
#define KRP   (3*H_ + 2*DP_) // 2368: padded rel  K (multiple of 4)

// ---------------------------------------------------------------------------
// Masked max-pool over S. One block per (b, n); mask row staged in LDS;
// 3 H-dims per thread. Matches (lhs*mask).max(S): accumulator starts at 0
// because unmasked positions contribute zeros. Entity variant also gathers
// size_emb[sum(mask)] into a stride-32 row (cols 25..31 zeroed).
// ---------------------------------------------------------------------------
__global__ __launch_bounds__(256) void pool_kernel(
    const float* __restrict__ lhs, const int* __restrict__ mask,
    const float* __restrict__ size_emb, float* __restrict__ pooled,
    float* __restrict__ esize_out) {
  const int bn  = blockIdx.x;
  const int b   = bn / E_;
  const int tid = threadIdx.x;

  __shared__ int smask[S_];
  __shared__ int slen;
  smask[tid] = mask[(size_t)bn * S_ + tid];
  __syncthreads();
  if (tid == 0) {
    int l = 0;
    for (int s = 0; s < S_; ++s) l += smask[s];
    slen = l;
  }

  const float* base = lhs + (size_t)b * S_ * H_;
  float a0 = 0.f, a1 = 0.f, a2 = 0.f;
  for (int s = 0; s < S_; ++s) {
    if (smask[s]) {
      const float* row = base + (size_t)s * H_;
      a0 = fmaxf(a0, row[tid]);
      a1 = fmaxf(a1, row[tid + 256]);
      a2 = fmaxf(a2, row[tid + 512]);
    }
  }
  pooled[(size_t)bn * H_ + tid]       = a0;
  pooled[(size_t)bn * H_ + tid + 256] = a1;
  pooled[(size_t)bn * H_ + tid + 512] = a2;

  __syncthreads();
  if (esize_out != nullptr && tid < DP_) {
    esize_out[(size_t)bn * DP_ + tid] =
        (tid < D_) ? size_emb[(size_t)slen * D_ + tid] : 0.f;
  }
}

// ---------------------------------------------------------------------------
// Weight packing into zero-padded [16][K_pad] layouts so the GEMM B-fragment
// loads are unconditional b64s (rows n>=valid and pad columns are zero).
// ---------------------------------------------------------------------------
__global__ __launch_bounds__(256) void pack_span_w_kernel(
    const float* __restrict__ w, float* __restrict__ wp) {
  const int i = blockIdx.x * 256 + threadIdx.x;   // 16*KSP
  if (i >= 16 * KSP) return;
  const int n = i / KSP, k = i % KSP;
  float v = 0.f;
  if (n < NET) {
    if (k < 2 * H_)                 v = w[(size_t)n * KSPAN + k];
    else if (k - 2 * H_ < D_)       v = w[(size_t)n * KSPAN + k];  // size cols
  }
  wp[i] = v;
}

__global__ __launch_bounds__(256) void pack_rel_w_kernel(
    const float* __restrict__ w, float* __restrict__ wp) {
  const int i = blockIdx.x * 256 + threadIdx.x;   // 16*KRP
  if (i >= 16 * KRP) return;
  const int n = i / KRP, k = i % KRP;
  float v = 0.f;
  if (n < NRT) {
    if (k < 3 * H_) {
      v = w[(size_t)n * KREL + k];
    } else if (k < 3 * H_ + DP_) {              // size0 pad block
      const int t = k - 3 * H_;
      if (t < D_) v = w[(size_t)n * KREL + 3 * H_ + t];
    } else {                                    // size1 pad block
      const int t = k - 3 * H_ - DP_;
      if (t < D_) v = w[(size_t)n * KREL + 3 * H_ + D_ + t];
    }
  }
  wp[i] = v;
}

// ---------------------------------------------------------------------------
// Branch-free K-segment: unconditional 8B-aligned b64 loads feeding
// V_WMMA_F32_16X16X4_F32. khalf in {0,2}, all bases even-float aligned.
// ---------------------------------------------------------------------------
__device__ __forceinline__ v8f gemm_seg(v8f acc, const float* __restrict__ pa,
                                        const float* __restrict__ pw,
                                        int klen, int khalf) {
#pragma unroll 4
  for (int k0 = 0; k0 < klen; k0 += 4) {
    const v2f a = *(const v2f*)(pa + k0 + khalf);
    const v2f w = *(const v2f*)(pw + k0 + khalf);
    acc = __builtin_amdgcn_wmma_f32_16x16x4_f32(
        false, a, false, w, (short)0, acc, false, false);
  }
  return acc;
}

// ---------------------------------------------------------------------------
// Entity logits: M=800, N=16 (9 valid), K=1568 padded. 2 waves/block,
// 25 blocks -> exactly 50 sixteen-row tiles, no guards.
// A 16x4 f32 layout: lanes 0-15 -> M, VGPR0/1 = K{0,1} (lo half) / K{2,3}
// (hi half); B mirrored with N on lanes; D: vgpr r -> M=r / M=8+r.
// ---------------------------------------------------------------------------
__global__ __launch_bounds__(64) void span_logit_kernel(
    const float* __restrict__ entity, const float* __restrict__ lhs,
    const float* __restrict__ esize, const float* __restrict__ span_wp,
    const float* __restrict__ span_b, float* __restrict__ out) {
  const int lane  = threadIdx.x & 31;
  const int tile  = blockIdx.x * 2 + (threadIdx.x >> 5);
  const int mrow  = tile * 16 + (lane & 15);
  const int b     = mrow / E_;
  const int n     = lane & 15;
  const int khalf = (lane >> 4) * 2;

  const float* pe = entity + (size_t)mrow * H_;
  const float* pc = lhs + (size_t)b * S_ * H_;     // s=0 row is the context
  const float* ps = esize + (size_t)mrow * DP_;
  const float* pw = span_wp + (size_t)n * KSP;

  v8f acc = {};
  acc = gemm_seg(acc, pe, pw,            H_,  khalf);  // pooled entity
  acc = gemm_seg(acc, pc, pw + H_,       H_,  khalf);  // context
  acc = gemm_seg(acc, ps, pw + 2 * H_,   DP_, khalf);  // size emb (padded)

  if (n < NET) {
    const float bias  = span_b[n];
    const int   mbase = tile * 16 + ((lane >> 4) ? 8 : 0);
#pragma unroll
    for (int r = 0; r < 8; ++r)
      out[(size_t)(mbase + r) * NET + n] = acc[r] + bias;
  }
}

// ---------------------------------------------------------------------------
// Relation logits: M=800, N=16 (5 valid), K=2368 padded, same scheme.
// ---------------------------------------------------------------------------
__global__ __launch_bounds__(64) void rel_logit_kernel(
    const float* __restrict__ relctx, const float* __restrict__ entity,
    const float* __restrict__ esize, const int* __restrict__ relations,
    const float* __restrict__ rel_wp, const float* __restrict__ rel_b,
    float* __restrict__ out) {
  const int lane  = threadIdx.x & 31;
  const int tile  = blockIdx.x * 2 + (threadIdx.x >> 5);
  const int mrow  = tile * 16 + (lane & 15);
  const int b     = mrow / R_;
  const int n     = lane & 15;
  const int khalf = (lane >> 4) * 2;

  const int e0 = relations[(size_t)mrow * 2 + 0];
  const int e1 = relations[(size_t)mrow * 2 + 1];
  const float* pr  = relctx + (size_t)mrow * H_;
  const float* pe0 = entity + ((size_t)b * E_ + e0) * H_;
  const float* pe1 = entity + ((size_t)b * E_ + e1) * H_;
  const float* ps0 = esize  + ((size_t)b * E_ + e0) * DP_;
  const float* ps1 = esize  + ((size_t)b * E_ + e1) * DP_;
  const float* pw  = rel_wp + (size_t)n * KRP;

  v8f acc = {};
  acc = gemm_seg(acc, pr,  pw,                 H_,  khalf);  // rel context
  acc = gemm_seg(acc, pe0, pw + H_,            H_,  khalf);  // head entity
  acc = gemm_seg(acc, pe1, pw + 2 * H_,        H_,  khalf);  // tail entity
  acc = gemm_seg(acc, ps0, pw + 3 * H_,        DP_, khalf);  // head size
  acc = gemm_seg(acc, ps1, pw + 3 * H_ + DP_,  DP_, khalf);  // tail size

  if (n < NRT) {
    const float bias  = rel_b[n];
    const int   mbase = tile * 16 + ((lane >> 4) ? 8 : 0);
#pragma unroll
    for (int r = 0; r < 8; ++r)
      out[(size_t)(mbase + r) * NRT + n] = acc[r] + bias;
  }
}

// ---------------------------------------------------------------------------
extern "C" void kernel_launch(void* const* d_in, const int* in_sizes, int n_in,
                              void* d_out, int out_size, void* d_ws, size_t ws_size,
                              hipStream_t stream) {
  const float* lhs       = (const float*)d_in[0];  // (B,S,H)
  const int*   ent_mask  = (const int*)  d_in[1];  // (B,E,S)
  const int*   relations = (const int*)  d_in[2];  // (B,R,2)
  const int*   rel_mask  = (const int*)  d_in[3];  // (B,R,S)
  const float* size_emb  = (const float*)d_in[4];  // (100,D)
  const float* span_w    = (const float*)d_in[5];  // (9, KSPAN)
  const float* span_b    = (const float*)d_in[6];  // (9,)
  const float* rel_w     = (const float*)d_in[7];  // (5, KREL)
  const float* rel_b     = (const float*)d_in[8];  // (5,)

  float* ws      = (float*)d_ws;
  float* ent     = ws;                               // 800*768
  float* relctx  = ent    + (size_t)B_ * E_ * H_;    // 800*768
  float* esz     = relctx + (size_t)B_ * R_ * H_;    // 800*32
  float* span_wp = esz    + (size_t)B_ * E_ * DP_;   // 16*1568
  float* rel_wp  = span_wp + (size_t)16 * KSP;       // 16*2368

  float* ent_logit = (float*)d_out;                          // 4*200*9
  float* rel_logit = (float*)d_out + (size_t)B_ * E_ * NET;  // 4*200*5

  // 0) pack weights into zero-padded [16][K_pad] buffers
  pack_span_w_kernel<<<(16 * KSP + 255) / 256, 256, 0, stream>>>(span_w, span_wp);
  pack_rel_w_kernel<<<(16 * KRP + 255) / 256, 256, 0, stream>>>(rel_w, rel_wp);

  // 1) masked max-pools (+ padded size-embedding gather for entities)
  pool_kernel<<<B_ * E_, 256, 0, stream>>>(lhs, ent_mask, size_emb, ent, esz);
  pool_kernel<<<B_ * R_, 256, 0, stream>>>(lhs, rel_mask, size_emb, relctx, nullptr);

  // 2) branch-free WMMA logit GEMMs: 50 tiles, 2 waves/block
  span_logit_kernel<<<25, 64, 0, stream>>>(ent, lhs, esz, span_wp, span_b, ent_logit);
  rel_logit_kernel<<<25, 64, 0, stream>>>(relctx, ent, esz, relations, rel_wp,
                                          rel_b, rel_logit);
}